// LightGCN_79671643341520
// MI455X (gfx1250) — compile-verified
//
#include <hip/hip_runtime.h>
#include <hip/hip_bf16.h>
#include <stdint.h>

#define NUM_USERS_C 50000
#define NUM_ITEMS_C 25000
#define N_NODES_C   (NUM_USERS_C + NUM_ITEMS_C)
#define DIM_C       64
#define NODE_ELEMS  ((size_t)N_NODES_C * DIM_C)   // 4,800,000 floats per buffer

typedef float v2f __attribute__((ext_vector_type(2)));
typedef float v8f __attribute__((ext_vector_type(8)));

// ---------------------------------------------------------------------------
// init: cur = concat(user_emb, item_emb); acc = cur; next = 0
// ---------------------------------------------------------------------------
__global__ __launch_bounds__(256) void k_init(const float* __restrict__ ue,
                                              const float* __restrict__ ie,
                                              float* __restrict__ cur,
                                              float* __restrict__ acc,
                                              float* __restrict__ nxt,
                                              int userElems, int totalElems) {
  int i = blockIdx.x * blockDim.x + threadIdx.x;
  if (i < totalElems) {
    float v = (i < userElems) ? ue[i] : ie[i - userElems];
    cur[i] = v;
    acc[i] = v;
    nxt[i] = 0.0f;
  }
}

// ---------------------------------------------------------------------------
// SpMM scatter: one wave32 per edge, each lane owns 2 of the 64 dims.
// Row gather is one contiguous 256B transaction per wave (L2-resident on
// MI455X: whole embedding set fits in 192MB L2). Scatter via f32 L2 atomics.
// Edge streams are one-touch -> prefetch ahead (global_prefetch_b8).
// ---------------------------------------------------------------------------
__global__ __launch_bounds__(256) void k_spmm(const float* __restrict__ cur,
                                              float* __restrict__ nxt,
                                              const int* __restrict__ esrc,
                                              const int* __restrict__ edst,
                                              const float* __restrict__ eval,
                                              int nEdges) {
  int t = blockIdx.x * blockDim.x + threadIdx.x;
  int e = t >> 5;
  int lane = t & 31;
  if (e >= nEdges) return;

  // stream-ahead prefetch of the edge arrays (one cacheline covers many edges)
  int pe = e + 4096;
  if (pe < nEdges) {
    __builtin_prefetch(&esrc[pe], 0, 0);
    __builtin_prefetch(&edst[pe], 0, 0);
    __builtin_prefetch(&eval[pe], 0, 0);
  }

  int s = esrc[e];
  int d = edst[e];
  float w = eval[e];

  const v2f* __restrict__ srow = (const v2f*)(cur + (size_t)s * DIM_C);
  v2f m = srow[lane];

  float* drow = nxt + (size_t)d * DIM_C + lane * 2;
  atomicAdd(drow + 0, w * m.x);
  atomicAdd(drow + 1, w * m.y);
}

// ---------------------------------------------------------------------------
// acc += just-produced layer; zero the other buffer so it can be the next
// scatter target (fused to save a pass).
// ---------------------------------------------------------------------------
__global__ __launch_bounds__(256) void k_acc_zero(float* __restrict__ acc,
                                                  const float* __restrict__ layer,
                                                  float* __restrict__ zbuf,
                                                  int totalElems) {
  int i = blockIdx.x * blockDim.x + threadIdx.x;
  if (i < totalElems) {
    acc[i] += layer[i];
    zbuf[i] = 0.0f;
  }
}

// ---------------------------------------------------------------------------
// gamma[b] = dot(acc_user[users[b]], acc_item[items[b]]) / 16
// One wave per 16 batch elements via V_WMMA_F32_16X16X4_F32:
//   A (16x4, M=batch-user rows, K=dim slice), B (4x16, K x batch-item cols)
//   accumulate 16 K-steps covering DIM=64; take diagonal of the 16x16 result.
// f32 A layout: lanes 0-15 -> M=lane, K={4kk+0,4kk+1}; lanes 16-31 -> M=lane-16,
// K={4kk+2,4kk+3}.  B mirrors with N in place of M.  EXEC stays all-ones
// (exact tiling: batch is a multiple of 16 and waves never diverge mid-tile).
// ---------------------------------------------------------------------------
__global__ __launch_bounds__(256) void k_gamma(const float* __restrict__ acc,
                                               const int* __restrict__ users,
                                               const int* __restrict__ items,
                                               float* __restrict__ out,
                                               int batch) {
  int t = blockIdx.x * blockDim.x + threadIdx.x;
  int wave = t >> 5;
  int lane = t & 31;
  int b0 = wave * 16;
  if (b0 >= batch) return;           // whole-wave uniform: EXEC stays full

  int half = lane >> 4;              // 0 -> K pair {0,1}; 1 -> K pair {2,3}
  int idx = lane & 15;               // M (for A) / N (for B) within the tile

  size_t urow = (size_t)users[b0 + idx] * DIM_C;
  size_t irow = ((size_t)NUM_USERS_C + (size_t)items[b0 + idx]) * DIM_C;
  const float* up = acc + urow + 2 * half;
  const float* ip = acc + irow + 2 * half;

  v8f c = {};
#pragma unroll
  for (int kk = 0; kk < 16; ++kk) {
    v2f a = *(const v2f*)(up + 4 * kk);
    v2f b = *(const v2f*)(ip + 4 * kk);
    // 8 args: (neg_a, A, neg_b, B, c_mod, C, reuse_a, reuse_b)
    c = __builtin_amdgcn_wmma_f32_16x16x4_f32(false, a, false, b,
                                              (short)0, c, false, false);
  }

  // Diagonal of D: m<8 -> lane m, vgpr m ; m>=8 -> lane m+16, vgpr m-8
  int r = -1;
  if (lane < 8)        r = lane;         // writes gamma[b0 + lane]
  else if (lane >= 24) r = lane - 24;    // writes gamma[b0 + lane - 16]
  if (r >= 0) {
    float g = 0.0f;
#pragma unroll
    for (int j = 0; j < 8; ++j)
      if (r == j) g = c[j];
    int b = (lane < 8) ? (b0 + lane) : (b0 + lane - 16);
    // light_out = acc / (N_LAYERS+1); gamma = dot(u,i) -> scale by 1/16
    out[b] = g * 0.0625f;
  }
}

// ---------------------------------------------------------------------------
extern "C" void kernel_launch(void* const* d_in, const int* in_sizes, int n_in,
                              void* d_out, int out_size, void* d_ws, size_t ws_size,
                              hipStream_t stream) {
  const float* user_emb = (const float*)d_in[0];
  const float* item_emb = (const float*)d_in[1];
  const int*   edge_src = (const int*)d_in[2];
  const int*   edge_dst = (const int*)d_in[3];
  const float* edge_val = (const float*)d_in[4];
  const int*   users    = (const int*)d_in[5];
  const int*   items    = (const int*)d_in[6];
  float*       out      = (float*)d_out;

  const int nEdges    = in_sizes[2];
  const int batch     = in_sizes[5];
  const int userElems = in_sizes[0];
  const int totalElems = (int)NODE_ELEMS;

  float* ws  = (float*)d_ws;
  float* cur = ws;
  float* nxt = ws + NODE_ELEMS;
  float* acc = ws + 2 * NODE_ELEMS;

  const int TPB = 256;
  dim3 gridElems((totalElems + TPB - 1) / TPB);
  k_init<<<gridElems, TPB, 0, stream>>>(user_emb, item_emb, cur, acc, nxt,
                                        userElems, totalElems);

  // 3 propagation layers: scatter cur -> nxt, acc += nxt, zero old cur, swap
  long long spmmThreads = (long long)nEdges * 32;
  dim3 gridSpmm((unsigned)((spmmThreads + TPB - 1) / TPB));
  for (int layer = 0; layer < 3; ++layer) {
    k_spmm<<<gridSpmm, TPB, 0, stream>>>(cur, nxt, edge_src, edge_dst,
                                         edge_val, nEdges);
    k_acc_zero<<<gridElems, TPB, 0, stream>>>(acc, nxt, cur, totalElems);
    float* tmp = cur; cur = nxt; nxt = tmp;
  }

  // final batched dot via WMMA (16 batch elems per wave)
  int tiles = (batch + 15) / 16;
  long long gThreads = (long long)tiles * 32;
  dim3 gridGamma((unsigned)((gThreads + TPB - 1) / TPB));
  k_gamma<<<gridGamma, TPB, 0, stream>>>(acc, users, items, out, batch);
}